// SetMatchingLoss_66202625900688
// MI455X (gfx1250) — compile-verified
//
#include <hip/hip_runtime.h>
#include <hip/hip_bf16.h>
#include <math.h>

// ---------------------------------------------------------------------------
// SetMatchingLoss for MI455X (gfx1250, wave32)
//   logits: (M, 4, 101) f32   target: (M, 4) int   weight: (101,) f32
//   out: scalar f32 = mean over rows of min-over-24-perms assignment cost
//
// Memory-bound (212 MB streamed @ 23.3 TB/s ~ 9.1 us). Data path:
//   TDM (tensor_load_to_lds) : 50.5 KB chunk (32 rows) -> LDS per block
//   wave32 shuffles          : softmax reductions + 4x4 assignment min
// ---------------------------------------------------------------------------

#define NUM_CLASSES   101
#define ROW_FLOATS    404            // 4 * 101
#define ROWS_PER_BLK  32
#define CHUNK_FLOATS  (ROWS_PER_BLK * ROW_FLOATS)   // 12928 floats = 51712 B
#define THREADS       256

typedef unsigned int u32;
typedef __attribute__((ext_vector_type(4))) u32 tdm_v4u;
typedef __attribute__((ext_vector_type(8))) int tdm_v8i;
typedef __attribute__((ext_vector_type(4))) int tdm_v4i;

#if __has_builtin(__builtin_amdgcn_tensor_load_to_lds) && \
    __has_builtin(__builtin_amdgcn_s_wait_tensorcnt)
#define USE_TDM 1
#else
#define USE_TDM 0
#endif

// 24 permutations of {0,1,2,3}; byte k = perm[k]
__device__ __constant__ u32 PERMS[24] = {
    0x03020100u, 0x02030100u, 0x03010200u, 0x01030200u, 0x02010300u, 0x01020300u,
    0x03020001u, 0x02030001u, 0x03000201u, 0x00030201u, 0x02000301u, 0x00020301u,
    0x03010002u, 0x01030002u, 0x03000102u, 0x00030102u, 0x01000302u, 0x00010302u,
    0x02010003u, 0x01020003u, 0x02000103u, 0x00020103u, 0x01000203u, 0x00010203u
};

__device__ __forceinline__ float wave_max(float v) {
    #pragma unroll
    for (int m = 16; m > 0; m >>= 1) v = fmaxf(v, __shfl_xor(v, m, 32));
    return v;
}
__device__ __forceinline__ float wave_sum(float v) {
    #pragma unroll
    for (int m = 16; m > 0; m >>= 1) v += __shfl_xor(v, m, 32);
    return v;
}
__device__ __forceinline__ float wave_min(float v) {
    #pragma unroll
    for (int m = 16; m > 0; m >>= 1) v = fminf(v, __shfl_xor(v, m, 32));
    return v;
}

__global__ __launch_bounds__(THREADS)
void set_match_loss_stage1(const float* __restrict__ logits,
                           const int*   __restrict__ target,
                           const float* __restrict__ weight,
                           float*       __restrict__ partial) {
    __shared__ float smem[CHUNK_FLOATS];
    __shared__ float wsum[THREADS / 32];

    const int  blk   = blockIdx.x;
    const float* gsrc = logits + (size_t)blk * CHUNK_FLOATS;

#if USE_TDM
    if (threadIdx.x < 32) {
        // ---- Tensor DMA descriptor: 1 row of 12928 fp32, global -> LDS ----
        unsigned long long ga = (unsigned long long)(const void*)gsrc;
        tdm_v4u g0;
        g0.x = 1u;                                      // count=1, user mode
        g0.y = (u32)(uintptr_t)(void*)&smem[0];         // lds_addr (bytes)
        g0.z = (u32)(ga & 0xffffffffu);                 // global_addr[31:0]
        g0.w = (u32)((ga >> 32) & 0x1ffffffu)           // global_addr[56:32]
             | (2u << 30);                              // type = 2 (image)

        tdm_v8i g1;
        g1[0] = (int)(2u << 16);                        // data_size = 4 B
        g1[1] = (int)((CHUNK_FLOATS & 0xffffu) << 16);  // tensor_dim0[15:0]
        g1[2] = (int)(((CHUNK_FLOATS >> 16) & 0xffffu)  // tensor_dim0[31:16]
             | (1u << 16));                             // tensor_dim1 = 1
        g1[3] = (int)((CHUNK_FLOATS & 0xffffu) << 16);  // tile_dim0 = 12928
        g1[4] = 1;                                      // tile_dim1=1, tile_dim2=0
        g1[5] = (int)CHUNK_FLOATS;                      // tensor_dim0_stride lo
        g1[6] = 0;
        g1[7] = 0;

        tdm_v4i gz = {0, 0, 0, 0};
#if defined(__clang_major__) && (__clang_major__ >= 23)
        tdm_v8i gz8 = {0, 0, 0, 0, 0, 0, 0, 0};
        __builtin_amdgcn_tensor_load_to_lds(g0, g1, gz, gz, gz8, 0);
#else
        __builtin_amdgcn_tensor_load_to_lds(g0, g1, gz, gz, 0);
#endif
        __builtin_amdgcn_s_wait_tensorcnt(0);
    }
#else
    for (int i = threadIdx.x; i < CHUNK_FLOATS / 4; i += THREADS)
        ((float4*)smem)[i] = ((const float4*)gsrc)[i];
#endif
    __syncthreads();

    const int lane = threadIdx.x & 31;
    const int wave = threadIdx.x >> 5;

    float acc = 0.0f;
    #pragma unroll
    for (int rr = 0; rr < ROWS_PER_BLK / (THREADS / 32); ++rr) {
        const int r    = wave * (ROWS_PER_BLK / (THREADS / 32)) + rr;
        const int grow = blk * ROWS_PER_BLK + r;
        const float* row = smem + r * ROW_FLOATS;

        // --- log-sum-exp per slot (101 classes, 32 lanes, 4 strided reads) ---
        float lse[4];
        #pragma unroll
        for (int k = 0; k < 4; ++k) {
            const float* b = row + k * NUM_CLASSES;
            float x0 = b[lane];
            float x1 = b[lane + 32];
            float x2 = b[lane + 64];
            float x3 = (lane < NUM_CLASSES - 96) ? b[lane + 96] : -INFINITY;
            float mx = wave_max(fmaxf(fmaxf(x0, x1), fmaxf(x2, x3)));
            float s  = __expf(x0 - mx) + __expf(x1 - mx) + __expf(x2 - mx);
            if (lane < NUM_CLASSES - 96) s += __expf(x3 - mx);
            s = wave_sum(s);
            lse[k] = mx + __logf(s);
        }

        // --- cost matrix on lanes 0..15: lane = 4*k + j ---
        float c = 0.0f;
        if (lane < 16) {
            const int k = lane >> 2, j = lane & 3;
            const int t = target[(size_t)grow * 4 + j];
            const float w = weight[t];
            c = -w * (row[k * NUM_CLASSES + t] - lse[k]);
        }

        // --- lanes 0..23 each evaluate one permutation ---
        float best;
        if (lane < 24) {
            const u32 p = PERMS[lane];
            float s0 = __shfl(c,      (int)( p        & 3u), 32);
            float s1 = __shfl(c,  4 + (int)((p >>  8) & 3u), 32);
            float s2 = __shfl(c,  8 + (int)((p >> 16) & 3u), 32);
            float s3 = __shfl(c, 12 + (int)((p >> 24) & 3u), 32);
            best = s0 + s1 + s2 + s3;
        } else {
            best = INFINITY;
        }
        acc += wave_min(best);
    }

    if (lane == 0) wsum[wave] = acc;
    __syncthreads();
    if (threadIdx.x == 0) {
        float t = 0.0f;
        #pragma unroll
        for (int i = 0; i < THREADS / 32; ++i) t += wsum[i];   // fixed order
        partial[blk] = t;
    }
}

__global__ __launch_bounds__(THREADS)
void set_match_loss_stage2(const float* __restrict__ partial,
                           float* __restrict__ out, int n, float inv_m) {
    __shared__ float red[THREADS];
    float a = 0.0f;
    for (int i = threadIdx.x; i < n; i += THREADS) a += partial[i];  // fixed order
    red[threadIdx.x] = a;
    __syncthreads();
    for (int s = THREADS / 2; s > 0; s >>= 1) {
        if (threadIdx.x < s) red[threadIdx.x] += red[threadIdx.x + s];
        __syncthreads();
    }
    if (threadIdx.x == 0) out[0] = red[0] * inv_m;
}

extern "C" void kernel_launch(void* const* d_in, const int* in_sizes, int n_in,
                              void* d_out, int out_size, void* d_ws, size_t ws_size,
                              hipStream_t stream) {
    const float* logits = (const float*)d_in[0];
    const int*   target = (const int*)d_in[1];
    const float* weight = (const float*)d_in[2];
    float* out = (float*)d_out;
    float* partial = (float*)d_ws;

    const int rows    = in_sizes[0] / ROW_FLOATS;     // 131072
    const int nblocks = rows / ROWS_PER_BLK;          // 4096

    set_match_loss_stage1<<<nblocks, THREADS, 0, stream>>>(logits, target, weight, partial);
    set_match_loss_stage2<<<1, THREADS, 0, stream>>>(partial, out, nblocks, 1.0f / (float)rows);
}